// HausdorffMetric_65781719106153
// MI455X (gfx1250) — compile-verified
//
#include <hip/hip_runtime.h>
#include <stdint.h>

// Problem geometry (fixed by reference setup_inputs)
#define B_  2
#define D_  64
#define H_  128
#define W_  128
#define VOL  (D_*H_*W_)          // 1048576
#define NVOX (B_*VOL)            // 2097152
#define BIGF 1e10f
#define MAXSQ (63*63 + 127*127 + 127*127)   // 36227
#define NBINS (MAXSQ + 1)                    // 36228

// ---------------------------------------------------------------------------
// K1: threshold -> masks -> cross-erosion edges; seed squared-EDT fields.
//   d2gt  seeded with gt edges  (0 at edge, BIG elsewhere)
//   d2pred seeded with pred edges
// ---------------------------------------------------------------------------
__global__ __launch_bounds__(256)
void edges_init_kernel(const float* __restrict__ pred,
                       const float* __restrict__ targ,
                       uint8_t* __restrict__ ep, uint8_t* __restrict__ eg,
                       float* __restrict__ d2gt, float* __restrict__ d2pred)
{
    int v = blockIdx.x * blockDim.x + threadIdx.x;
    if (v >= NVOX) return;
    int x = v % W_;
    int y = (v / W_) % H_;
    int z = (v / (W_ * H_)) % D_;

    auto nb = [&](const float* p, int dz, int dy, int dx) -> bool {
        int zz = z + dz, yy = y + dy, xx = x + dx;
        if (zz < 0 || zz >= D_ || yy < 0 || yy >= H_ || xx < 0 || xx >= W_)
            return false; // scipy border_value=0
        return p[v + dz * (W_ * H_) + dy * W_ + dx] > 0.5f;
    };

    bool mp = pred[v] > 0.5f;
    bool mt = targ[v] > 0.5f;

    bool e_p = false, e_g = false;
    if (mp) {
        bool er = nb(pred,-1,0,0) && nb(pred,1,0,0) &&
                  nb(pred,0,-1,0) && nb(pred,0,1,0) &&
                  nb(pred,0,0,-1) && nb(pred,0,0,1);
        e_p = !er;
    }
    if (mt) {
        bool er = nb(targ,-1,0,0) && nb(targ,1,0,0) &&
                  nb(targ,0,-1,0) && nb(targ,0,1,0) &&
                  nb(targ,0,0,-1) && nb(targ,0,0,1);
        e_g = !er;
    }

    ep[v] = (uint8_t)e_p;
    eg[v] = (uint8_t)e_g;
    d2gt[v]   = e_g ? 0.0f : BIGF;
    d2pred[v] = e_p ? 0.0f : BIGF;
}

// ---------------------------------------------------------------------------
// K2: one EDT parabola min-convolution pass along one axis.
// One workgroup per line; the line is staged in LDS via the CDNA5 async
// global->LDS path (global_load_async_to_lds_b32, ASYNCcnt), then each lane
// computes min_j line[j] + (t-j)^2 (broadcast LDS reads, no bank conflicts).
// Line base address: base = (l / G) * GS + (l % G) * IS, element stride `step`.
// grid.y selects which distance field (0 -> d2gt, 1 -> d2pred).
// ---------------------------------------------------------------------------
__global__ __launch_bounds__(128)
void edt_pass_kernel(float* __restrict__ f0, float* __restrict__ f1,
                     int L, int step, int G, int GS, int IS)
{
    float* f = (blockIdx.y == 0) ? f0 : f1;
    int l = blockIdx.x;
    int base = (l / G) * GS + (l % G) * IS;
    int t = threadIdx.x;                    // t in [0, L)

    __shared__ float line[128];

    const float* gptr = f + base + t * step;
    // LDS byte address: low 32 bits of the flat pointer to a __shared__ object
    // are the LDS offset (aperture mapping discards the upper bits).
    unsigned lds_off = (unsigned)(uintptr_t)(&line[t]);

    // CDNA5 async copy global -> LDS (tracked with ASYNCcnt)
    asm volatile("global_load_async_to_lds_b32 %0, %1, off"
                 :: "v"(lds_off), "v"(gptr) : "memory");
    asm volatile("s_wait_asynccnt 0" ::: "memory");
    __syncthreads();

    float ti = (float)t;
    float m = BIGF + (float)MAXSQ; // > any reachable value; min over j includes j=t
    #pragma unroll 4
    for (int j = 0; j < L; ++j) {
        float dj = ti - (float)j;
        m = fminf(m, line[j] + dj * dj);
    }
    f[base + t * step] = m;
}

// ---------------------------------------------------------------------------
// K3: zero histograms (ws is poisoned by the harness; must clear every call)
// ---------------------------------------------------------------------------
__global__ __launch_bounds__(256)
void zero_hist_kernel(unsigned int* __restrict__ hist)
{
    int i = blockIdx.x * blockDim.x + threadIdx.x;
    if (i < B_ * 2 * NBINS) hist[i] = 0u;
}

// ---------------------------------------------------------------------------
// K4: exact histogram of integer squared distances at edge voxels.
//   pair index (b*2 + 0): pred edges vs distance-to-gt-edge  (h_pg)
//   pair index (b*2 + 1): gt edges   vs distance-to-pred-edge (h_gp)
// ---------------------------------------------------------------------------
__global__ __launch_bounds__(256)
void hist_kernel(const uint8_t* __restrict__ ep, const uint8_t* __restrict__ eg,
                 const float* __restrict__ d2gt, const float* __restrict__ d2pred,
                 unsigned int* __restrict__ hist)
{
    int v = blockIdx.x * blockDim.x + threadIdx.x;
    if (v >= NVOX) return;
    int b = v / VOL;
    if (ep[v]) {
        int q = (int)fminf(d2gt[v], (float)MAXSQ);   // exact small integers
        atomicAdd(&hist[(b * 2 + 0) * NBINS + q], 1u);
    }
    if (eg[v]) {
        int q = (int)fminf(d2pred[v], (float)MAXSQ);
        atomicAdd(&hist[(b * 2 + 1) * NBINS + q], 1u);
    }
}

// ---------------------------------------------------------------------------
// K5: percentile via cumulative histogram (exact order statistics), matching
// the reference fp32 interpolation, then out[b] = max(h_pg, h_gp).
// ---------------------------------------------------------------------------
__global__ __launch_bounds__(32)
void finalize_kernel(const unsigned int* __restrict__ hist, float* __restrict__ out)
{
    __shared__ float h[B_ * 2];
    int t = threadIdx.x;
    if (t < B_ * 2) {
        const unsigned int* hh = hist + t * NBINS;
        long long n = 0;
        for (int i = 0; i < NBINS; ++i) n += hh[i];

        float res;
        if (n == 0) {
            res = INFINITY;  // percentile of empty masked set -> inf (matches ref)
        } else {
            float nf   = (float)n;
            float pos  = 0.95f * (nf - 1.0f);          // fp32, as in reference
            float fl   = floorf(pos);
            long long lo = (long long)fl;
            if (lo < 0) lo = 0;
            if (lo > n - 1) lo = n - 1;
            long long hi = lo + 1;
            if (hi > n - 1) hi = n - 1;
            float frac = pos - fl;

            long long cum = 0;
            int sql = -1, sqh = -1;
            for (int i = 0; i < NBINS; ++i) {
                cum += hh[i];
                if (sql < 0 && cum > lo) sql = i;
                if (sqh < 0 && cum > hi) { sqh = i; break; }
            }
            float a = sqrtf((float)sql);
            float c = sqrtf((float)sqh);
            res = a + frac * (c - a);
        }
        h[t] = res;
    }
    __syncthreads();
    if (t < B_) out[t] = fmaxf(h[t * 2 + 0], h[t * 2 + 1]);  // out shape [B,1]
}

// ---------------------------------------------------------------------------
// Launch
// ---------------------------------------------------------------------------
extern "C" void kernel_launch(void* const* d_in, const int* in_sizes, int n_in,
                              void* d_out, int out_size, void* d_ws, size_t ws_size,
                              hipStream_t stream) {
    const float* pred = (const float*)d_in[0];
    const float* targ = (const float*)d_in[1];
    float* out = (float*)d_out;

    // Workspace layout (≈20.6 MB)
    float* d2gt   = (float*)d_ws;                               // NVOX f32
    float* d2pred = d2gt + NVOX;                                // NVOX f32
    unsigned int* hist = (unsigned int*)(d2pred + NVOX);        // B*2*NBINS u32
    uint8_t* ep = (uint8_t*)(hist + B_ * 2 * NBINS);            // NVOX u8
    uint8_t* eg = ep + NVOX;                                    // NVOX u8

    edges_init_kernel<<<(NVOX + 255) / 256, 256, 0, stream>>>(pred, targ, ep, eg, d2gt, d2pred);
    zero_hist_kernel<<<(B_ * 2 * NBINS + 255) / 256, 256, 0, stream>>>(hist);

    // EDT passes over both fields (grid.y = field). Axis order irrelevant
    // (exact separable EDT). base = (l/G)*GS + (l%G)*IS, stride `step`.
    // Axis W: lines (b,z,y,:)  L=128 step=1       base = l*W
    edt_pass_kernel<<<dim3(B_ * D_ * H_, 2), 128, 0, stream>>>(
        d2gt, d2pred, 128, 1, B_ * D_ * H_, 0, W_);
    // Axis H: lines (b,z,:,x)  L=128 step=W       base = (l/W)*H*W + l%W
    edt_pass_kernel<<<dim3(B_ * D_ * W_, 2), 128, 0, stream>>>(
        d2gt, d2pred, 128, W_, W_, H_ * W_, 1);
    // Axis D: lines (b,:,y,x)  L=64  step=H*W     base = (l/(H*W))*VOL + l%(H*W)
    edt_pass_kernel<<<dim3(B_ * H_ * W_, 2), 64, 0, stream>>>(
        d2gt, d2pred, 64, H_ * W_, H_ * W_, VOL, 1);

    hist_kernel<<<(NVOX + 255) / 256, 256, 0, stream>>>(ep, eg, d2gt, d2pred, hist);
    finalize_kernel<<<1, 32, 0, stream>>>(hist, out);
}